// RoIAlignRotated_23845658427804
// MI455X (gfx1250) — compile-verified
//
#include <hip/hip_runtime.h>
#include <math.h>

// ---------------------------------------------------------------------------
// RoIAlignRotated, fp32, B=2 C=256 H=200 W=200, R rois, 7x7 out, 2x2 samples.
// Strategy (MI455X): features (82MB) are L2-resident (192MB L2). Transpose
// once to NHWC so per-pixel channel gathers are contiguous 1KB reads
// (float4/lane, coalesced). Gather kernel: 1 block / roi, LDS-staged output
// for coalesced [R,C,49] stores. Uses gfx1250 async global->LDS copies
// (GLOBAL_LOAD_ASYNC_TO_LDS_B32 + s_wait_asynccnt) where available.
// ---------------------------------------------------------------------------

#ifndef __has_builtin
#define __has_builtin(x) 0
#endif

#if defined(__gfx1250__) && __has_builtin(__builtin_amdgcn_global_load_async_to_lds_b32)
#define USE_ASYNC_LDS 1
#endif

typedef __attribute__((address_space(1))) int as1_int;  // global
typedef __attribute__((address_space(3))) int as3_int;  // LDS

__device__ __forceinline__ void g2l_async_b32(const float* g, float* l) {
#ifdef USE_ASYNC_LDS
  __builtin_amdgcn_global_load_async_to_lds_b32(
      (as1_int*)(void*)g, (as3_int*)(void*)l, 0, 0);
#else
  *l = *g;
#endif
}

__device__ __forceinline__ void async_wait_all() {
#ifdef USE_ASYNC_LDS
#if __has_builtin(__builtin_amdgcn_s_wait_asynccnt)
  __builtin_amdgcn_s_wait_asynccnt(0);
#else
  asm volatile("s_wait_asynccnt 0" ::: "memory");
#endif
#endif
}

namespace cfg {
constexpr int   B = 2, C = 256, H = 200, W = 200;
constexpr int   OH = 7, OW = 7, G = 2;
constexpr int   NBIN = OH * OW;       // 49
constexpr int   LPITCH = 257;         // padded channel pitch (C+1) for LDS
constexpr float SCALE = 0.25f;
}

// ---------------------------------------------------------------------------
// Kernel 1: NCHW -> NHWC transpose. Per batch: [C, S=H*W] -> [S, C].
// 32x32 tiles via LDS, loads issued through the async global->LDS path.
// ---------------------------------------------------------------------------
__global__ __launch_bounds__(256) void nchw_to_nhwc_kernel(
    const float* __restrict__ feat, float* __restrict__ nhwc) {
  constexpr int S = cfg::H * cfg::W;    // 40000, multiple of 32
  constexpr int C = cfg::C;             // 256,   multiple of 32
  __shared__ float tile[32][33];

  const int b  = blockIdx.z;
  const int s0 = blockIdx.x * 32;
  const int c0 = blockIdx.y * 32;
  const int tx  = threadIdx.x & 31;
  const int ty0 = threadIdx.x >> 5;     // 0..7

  const float* src = feat + (size_t)b * C * S;
#pragma unroll
  for (int k = 0; k < 4; ++k) {
    const int cy = ty0 + k * 8;
    g2l_async_b32(src + (size_t)(c0 + cy) * S + (s0 + tx), &tile[cy][tx]);
  }
  async_wait_all();
  __syncthreads();

  float* dst = nhwc + (size_t)b * S * C;
#pragma unroll
  for (int k = 0; k < 4; ++k) {
    const int sy = ty0 + k * 8;
    dst[(size_t)(s0 + sy) * C + (c0 + tx)] = tile[tx][sy];  // stride-33 read: no bank conflicts
  }
}

// ---------------------------------------------------------------------------
// Shared per-sample geometry (matches the reference exactly).
// ---------------------------------------------------------------------------
struct RoiGeom {
  int   b;
  float cx, cy, bh, bw, nh2, nw2, st, ct;
};

__device__ __forceinline__ RoiGeom load_geom(const float* sroi) {
  RoiGeom g;
  g.b  = (int)sroi[0];
  g.cx = sroi[1] * cfg::SCALE - 0.5f;   // ALIGNED offset
  g.cy = sroi[2] * cfg::SCALE - 0.5f;
  const float rw = sroi[3] * cfg::SCALE;
  const float rh = sroi[4] * cfg::SCALE;
  const float th = sroi[5];             // CLOCKWISE == false
  g.bh  = rh * (1.0f / cfg::OH);
  g.bw  = rw * (1.0f / cfg::OW);
  g.nh2 = -0.5f * rh;
  g.nw2 = -0.5f * rw;
  g.st  = sinf(th);
  g.ct  = cosf(th);
  return g;
}

// Compute sample point; returns false if outside (-1,H)x(-1,W).
__device__ __forceinline__ bool sample_weights(
    const RoiGeom& g, int ph, int pw, int s,
    int& yl, int& yh, int& xl, int& xh,
    float& w1, float& w2, float& w3, float& w4) {
  const float iy = ((float)(s >> 1) + 0.5f) * (1.0f / cfg::G);
  const float ix = ((float)(s & 1)  + 0.5f) * (1.0f / cfg::G);
  const float yy = fmaf((float)ph + iy, g.bh, g.nh2);
  const float xx = fmaf((float)pw + ix, g.bw, g.nw2);
  float x = fmaf(yy, g.st, fmaf(xx,  g.ct, g.cx));
  float y = fmaf(yy, g.ct, fmaf(-xx, g.st, g.cy));
  if (!(y > -1.0f && y < (float)cfg::H && x > -1.0f && x < (float)cfg::W)) return false;
  y = fmaxf(y, 0.0f);
  x = fmaxf(x, 0.0f);
  yl = (int)y;  // y >= 0 -> trunc == floor
  xl = (int)x;
  float ly, lx;
  if (yl >= cfg::H - 1) { yl = yh = cfg::H - 1; ly = 0.0f; }
  else                  { yh = yl + 1; ly = y - (float)yl; }
  if (xl >= cfg::W - 1) { xl = xh = cfg::W - 1; lx = 0.0f; }
  else                  { xh = xl + 1; lx = x - (float)xl; }
  const float hy = 1.0f - ly, hx = 1.0f - lx;
  w1 = hy * hx; w2 = hy * lx; w3 = ly * hx; w4 = ly * lx;
  return true;
}

// ---------------------------------------------------------------------------
// Kernel 2a: gather from NHWC (fast path). 1 block per roi, 256 threads:
// thread = (bin-sub 0..3, channel-quad 0..63). float4 coalesced corner loads.
// ---------------------------------------------------------------------------
__global__ __launch_bounds__(256) void roi_align_rot_nhwc_kernel(
    const float* __restrict__ nhwc, const float* __restrict__ rois,
    float* __restrict__ out) {
  __shared__ float sroi[8];
  __shared__ float lbuf[cfg::NBIN * cfg::LPITCH];  // [bin][c], padded pitch 257

  const int r = blockIdx.x;
  const int t = threadIdx.x;

  if (t < 6) g2l_async_b32(rois + (size_t)r * 6 + t, &sroi[t]);
  async_wait_all();
  __syncthreads();

  const RoiGeom g = load_geom(sroi);
  const int sub = t >> 6;           // bin offset in group of 4
  const int c4  = (t & 63) << 2;    // channel quad start

  for (int grp = 0; grp < 13; ++grp) {
    const int bin = grp * 4 + sub;
    if (bin < cfg::NBIN) {
      const int ph = bin / cfg::OW;
      const int pw = bin - ph * cfg::OW;
      float4 acc = make_float4(0.f, 0.f, 0.f, 0.f);
#pragma unroll
      for (int s = 0; s < cfg::G * cfg::G; ++s) {
        int yl, yh, xl, xh; float w1, w2, w3, w4;
        if (sample_weights(g, ph, pw, s, yl, yh, xl, xh, w1, w2, w3, w4)) {
          const int rl = (g.b * cfg::H + yl) * cfg::W;
          const int rh = (g.b * cfg::H + yh) * cfg::W;
          const float4 v1 = *(const float4*)(nhwc + (size_t)(rl + xl) * cfg::C + c4);
          const float4 v2 = *(const float4*)(nhwc + (size_t)(rl + xh) * cfg::C + c4);
          const float4 v3 = *(const float4*)(nhwc + (size_t)(rh + xl) * cfg::C + c4);
          const float4 v4 = *(const float4*)(nhwc + (size_t)(rh + xh) * cfg::C + c4);
          acc.x += w1 * v1.x + w2 * v2.x + w3 * v3.x + w4 * v4.x;
          acc.y += w1 * v1.y + w2 * v2.y + w3 * v3.y + w4 * v4.y;
          acc.z += w1 * v1.z + w2 * v2.z + w3 * v3.z + w4 * v4.z;
          acc.w += w1 * v1.w + w2 * v2.w + w3 * v3.w + w4 * v4.w;
        }
      }
      const float inv = 1.0f / (cfg::G * cfg::G);
      float* lp = &lbuf[bin * cfg::LPITCH + c4];
      lp[0] = acc.x * inv; lp[1] = acc.y * inv; lp[2] = acc.z * inv; lp[3] = acc.w * inv;
    }
  }
  __syncthreads();

  // Coalesced writeback: out[r, c, bin] flat = r*C*49 + c*49 + bin
  float* obase = out + (size_t)r * cfg::C * cfg::NBIN;
  for (int i = t; i < cfg::C * cfg::NBIN; i += 256) {
    const int c   = i / cfg::NBIN;
    const int bin = i - c * cfg::NBIN;
    obase[i] = lbuf[bin * cfg::LPITCH + c];  // stride-257 read: conflict-free
  }
}

// ---------------------------------------------------------------------------
// Kernel 2b: fallback gather directly from NCHW (if workspace too small).
// thread = channel, one bin at a time; loads are strided but correct.
// ---------------------------------------------------------------------------
__global__ __launch_bounds__(256) void roi_align_rot_nchw_kernel(
    const float* __restrict__ feat, const float* __restrict__ rois,
    float* __restrict__ out) {
  __shared__ float sroi[8];
  __shared__ float lbuf[cfg::NBIN * cfg::LPITCH];

  const int r = blockIdx.x;
  const int t = threadIdx.x;  // channel

  if (t < 6) g2l_async_b32(rois + (size_t)r * 6 + t, &sroi[t]);
  async_wait_all();
  __syncthreads();

  const RoiGeom g = load_geom(sroi);
  const size_t cplane = ((size_t)g.b * cfg::C + t) * cfg::H * cfg::W;

  for (int bin = 0; bin < cfg::NBIN; ++bin) {
    const int ph = bin / cfg::OW;
    const int pw = bin - ph * cfg::OW;
    float acc = 0.0f;
#pragma unroll
    for (int s = 0; s < cfg::G * cfg::G; ++s) {
      int yl, yh, xl, xh; float w1, w2, w3, w4;
      if (sample_weights(g, ph, pw, s, yl, yh, xl, xh, w1, w2, w3, w4)) {
        const float v1 = feat[cplane + (size_t)yl * cfg::W + xl];
        const float v2 = feat[cplane + (size_t)yl * cfg::W + xh];
        const float v3 = feat[cplane + (size_t)yh * cfg::W + xl];
        const float v4 = feat[cplane + (size_t)yh * cfg::W + xh];
        acc += w1 * v1 + w2 * v2 + w3 * v3 + w4 * v4;
      }
    }
    lbuf[bin * cfg::LPITCH + t] = acc * (1.0f / (cfg::G * cfg::G));
  }
  __syncthreads();

  float* obase = out + (size_t)r * cfg::C * cfg::NBIN;
  for (int i = t; i < cfg::C * cfg::NBIN; i += 256) {
    const int c   = i / cfg::NBIN;
    const int bin = i - c * cfg::NBIN;
    obase[i] = lbuf[bin * cfg::LPITCH + c];
  }
}

// ---------------------------------------------------------------------------
// Host launcher
// ---------------------------------------------------------------------------
extern "C" void kernel_launch(void* const* d_in, const int* in_sizes, int n_in,
                              void* d_out, int out_size, void* d_ws, size_t ws_size,
                              hipStream_t stream) {
  (void)n_in; (void)out_size;
  const float* feat = (const float*)d_in[0];
  const float* rois = (const float*)d_in[1];
  float* out = (float*)d_out;

  const int R = in_sizes[1] / 6;
  if (R <= 0) return;

  const size_t nhwc_bytes =
      (size_t)cfg::B * cfg::H * cfg::W * cfg::C * sizeof(float);

  if (ws_size >= nhwc_bytes) {
    float* nhwc = (float*)d_ws;
    dim3 tg((cfg::H * cfg::W) / 32, cfg::C / 32, cfg::B);  // (1250, 8, 2)
    nchw_to_nhwc_kernel<<<tg, 256, 0, stream>>>(feat, nhwc);
    roi_align_rot_nhwc_kernel<<<dim3(R), 256, 0, stream>>>(nhwc, rois, out);
  } else {
    roi_align_rot_nchw_kernel<<<dim3(R), 256, 0, stream>>>(feat, rois, out);
  }
}